// MultiHeadAttentionXL_30683246362647
// MI455X (gfx1250) — compile-verified
//
#include <hip/hip_runtime.h>
#include <hip/hip_bf16.h>

typedef __attribute__((ext_vector_type(2))) float v2f;
typedef __attribute__((ext_vector_type(8))) float v8f;

// D = A(16x4) * B(4x16) + C, fp32 WMMA. Immediates must be literals -> macro.
#define WMMA_F32X4(A_, B_, C_) \
  __builtin_amdgcn_wmma_f32_16x16x4_f32(false, (A_), false, (B_), (short)0, (C_), false, false)

enum : int {
  nBS = 4, nCS = 1024, nPS = 1024, nDIN = 1024,
  nH = 16, nD = 64, nT = 2048, nHD = 1024, nR = 4096
};
#define ATTN_SCALE 0.125f  // 1/sqrt(64)

// ---------------------------------------------------------------------------
// Generic fp32 WMMA GEMM: C[M,N] = A[M,K] @ B[K,N].
// A rows may be split per "batch": row r -> b=r/Trows, t=r%Trows;
//   t <  split : A0 + (b*split + t)*K
//   t >= split : A1 + (b*(Trows-split) + (t-split))*K
// (set split == Trows for a plain contiguous A). One wave per block,
// each wave computes a 32x64 C tile. grid = (N/64, M/32).
// ---------------------------------------------------------------------------
__global__ __launch_bounds__(32) void gemm_wmma_f32(
    const float* __restrict__ A0, const float* __restrict__ A1,
    int Trows, int split, int Kdim,
    const float* __restrict__ B, int ldb,
    float* __restrict__ C, int ldc) {
  const int gm = blockIdx.y << 5;
  const int gn = blockIdx.x << 6;
  const int lane = threadIdx.x;
  const int lm = lane & 15;
  const int lk = (lane >> 4) << 1;

  const float* ra[2];
#pragma unroll
  for (int i = 0; i < 2; ++i) {
    int r = gm + (i << 4) + lm;
    int bb = r / Trows;
    int t = r - bb * Trows;
    ra[i] = (t < split)
                ? (A0 + ((size_t)bb * split + t) * (size_t)Kdim)
                : (A1 + ((size_t)bb * (Trows - split) + (t - split)) * (size_t)Kdim);
  }
  const float* bcol = B + gn + lm;

  v8f acc[2][4];
#pragma unroll
  for (int i = 0; i < 2; ++i)
#pragma unroll
    for (int j = 0; j < 4; ++j)
      acc[i][j] = (v8f){0.f, 0.f, 0.f, 0.f, 0.f, 0.f, 0.f, 0.f};

  for (int k0 = 0; k0 < Kdim; k0 += 4) {
    v2f a0 = *(const v2f*)(ra[0] + k0 + lk);
    v2f a1 = *(const v2f*)(ra[1] + k0 + lk);
    const float* bp = bcol + (size_t)(k0 + lk) * ldb;
#pragma unroll
    for (int j = 0; j < 4; ++j) {
      v2f bv;
      bv.x = bp[j * 16];
      bv.y = bp[(size_t)ldb + j * 16];
      acc[0][j] = WMMA_F32X4(a0, bv, acc[0][j]);
      acc[1][j] = WMMA_F32X4(a1, bv, acc[1][j]);
    }
  }

  const int mrow = gm + ((lane >> 4) << 3);
#pragma unroll
  for (int i = 0; i < 2; ++i)
#pragma unroll
    for (int j = 0; j < 4; ++j)
#pragma unroll
      for (int g = 0; g < 8; ++g)
        C[(size_t)(mrow + (i << 4) + g) * ldc + gn + (j << 4) + lm] = acc[i][j][g];
}

// ---------------------------------------------------------------------------
// Fused TXL attention for one (batch, head, 16-query tile).
// LDS: sc[16][2048] fp32 score strip.
//   A: content scores (q+u)K^T via WMMA      -> sc
//   B: rel-shifted position scores via WMMA in anti-diagonal coords, scatter-add
//   C: masked softmax (rowwise, 8 lanes/row)
//   D: out = P @ V via WMMA                  -> AWV
// grid = (CS/16, H, BS), block = 128 (4 waves), dyn LDS = 16*2048*4 B.
// ---------------------------------------------------------------------------
__global__ __launch_bounds__(128) void txl_attn_kernel(
    const float* __restrict__ Q, const float* __restrict__ Kt,
    const float* __restrict__ Vt, const float* __restrict__ Pt,
    const float* __restrict__ uB, const float* __restrict__ vB,
    float* __restrict__ AWV) {
  extern __shared__ float sc[];  // 16 * nT floats

  const int tid = threadIdx.x;
  const int wave = tid >> 5;
  const int lane = tid & 31;
  const int lm = lane & 15;
  const int lk = (lane >> 4) << 1;

  const int bz = blockIdx.z;
  const int h = blockIdx.y;
  const int i0 = blockIdx.x << 4;
  const int r0 = bz * nCS + i0;              // global query row base
  const int kmax = i0 + 16 + nPS;            // exclusive key bound (mult of 16)

  // ---- Phase A: content scores -------------------------------------------
  v2f qu[16];
  {
    const float* qrow = Q + (size_t)(r0 + lm) * nHD + h * nD;
    const float* ub = uB + h * nD;
#pragma unroll
    for (int ks = 0; ks < 16; ++ks) {
      int k = (ks << 2) + lk;
      v2f a = *(const v2f*)(qrow + k);
      a.x += ub[k];
      a.y += ub[k + 1];
      qu[ks] = a;
    }
  }
  for (int jt = wave << 4; jt < kmax; jt += 64) {
    const float* kb = Kt + (size_t)(bz * nT + jt + lm) * nHD + h * nD;
    v8f acc = (v8f){0.f, 0.f, 0.f, 0.f, 0.f, 0.f, 0.f, 0.f};
#pragma unroll
    for (int ks = 0; ks < 16; ++ks) {
      int k = (ks << 2) + lk;
      v2f bv = *(const v2f*)(kb + k);
      acc = WMMA_F32X4(qu[ks], bv, acc);
    }
#pragma unroll
    for (int g = 0; g < 8; ++g) {
      int m = ((lane >> 4) << 3) + g;
      sc[m * nT + jt + lm] = acc[g];
    }
  }
  __syncthreads();

  // ---- Phase B: rel-shifted position scores ------------------------------
  // Anti-diagonal coord s = j - r. Branch d(s): s<=-2048 -> d0 (c=s+4095, r1=r)
  // -2046<=s<=1 -> d1 (c=s+2046, r1=r+1); s>=3 -> d2 (c=s-3, r1=r+2).
  // s = -2047 and s = 2 are exact zeros (k1==0 in the reshape shift).
  {
    const int smax = nPS - bz * nCS;         // kept-mask bound: s <= 1024*(1-b)
    const int smin = -(r0 + 15);
    const int tLo = smin >> 4;               // arithmetic shift == floor/16
    const int tHi = smax >> 4;
    for (int ti = tLo + wave; ti <= tHi; ti += 4) {
      const int s0 = ti << 4;
      const int cat0 = (s0 <= -2048) ? 0 : ((s0 <= 2) ? 1 : 2);
      const int s15 = s0 + 15;
      const int cat1 = (s15 <= -2048) ? 0 : ((s15 <= 2) ? 1 : 2);
      for (int d = cat0; d <= cat1; ++d) {
        const int Coff = (d == 0) ? 4095 : ((d == 1) ? 2046 : -3);
        int r1 = r0 + lm + d;
        if (r1 > nR - 1) r1 = nR - 1;        // clamped load; never scattered
        int cN = s0 + lm + Coff;
        cN = (cN < 0) ? 0 : ((cN > nT - 1) ? nT - 1 : cN);
        const float* qb = Q + (size_t)r1 * nHD + h * nD;
        const float* vb = vB + h * nD;
        const float* pb = Pt + (size_t)cN * nHD + h * nD;
        v8f acc = (v8f){0.f, 0.f, 0.f, 0.f, 0.f, 0.f, 0.f, 0.f};
#pragma unroll
        for (int ks = 0; ks < 16; ++ks) {
          int k = (ks << 2) + lk;
          v2f a = *(const v2f*)(qb + k);
          a.x += vb[k];
          a.y += vb[k + 1];
          v2f bv = *(const v2f*)(pb + k);
          acc = WMMA_F32X4(a, bv, acc);
        }
        const int sC = s0 + lm;
        const bool okB = (d == 0) ? (sC <= -2048)
                                  : ((d == 1) ? (sC >= -2046 && sC <= 1)
                                              : (sC >= 3));
#pragma unroll
        for (int g = 0; g < 8; ++g) {
          int m = ((lane >> 4) << 3) + g;
          int j = sC + r0 + m;               // each (m,j) hit at most once
          if (okB && j >= 0 && j < kmax) sc[m * nT + j] += acc[g];
        }
      }
    }
  }
  __syncthreads();

  // ---- Phase C: masked softmax (8 lanes per row) -------------------------
  {
    const int row = tid >> 3;
    const int l8 = tid & 7;
    float* rp = sc + row * nT;
    const int keepEnd = i0 + row + nPS + 1;  // exclusive kept bound
    float mx = -3.0e38f;
    for (int j = l8; j < keepEnd; j += 8) mx = fmaxf(mx, rp[j]);
#pragma unroll
    for (int o = 4; o > 0; o >>= 1) mx = fmaxf(mx, __shfl_xor(mx, o, 8));
    float sum = 0.f;
    for (int j = l8; j < kmax; j += 8) {
      float e = 0.f;
      if (j < keepEnd) e = __expf((rp[j] - mx) * ATTN_SCALE);
      rp[j] = e;
      sum += e;
    }
#pragma unroll
    for (int o = 4; o > 0; o >>= 1) sum += __shfl_xor(sum, o, 8);
    const float inv = 1.0f / sum;
    for (int j = l8; j < kmax; j += 8) rp[j] *= inv;
  }
  __syncthreads();

  // ---- Phase D: out = P @ V (each wave one 16-wide D chunk) --------------
  {
    const int d0 = wave << 4;
    const float* vb = Vt + (size_t)(bz * nT) * nHD + h * nD + d0 + lm;
    v8f acc = (v8f){0.f, 0.f, 0.f, 0.f, 0.f, 0.f, 0.f, 0.f};
    for (int k0 = 0; k0 < kmax; k0 += 4) {
      int kk = k0 + lk;
      v2f a = *(const v2f*)(sc + lm * nT + kk);
      v2f bv;
      bv.x = vb[(size_t)kk * nHD];
      bv.y = vb[(size_t)(kk + 1) * nHD];
      acc = WMMA_F32X4(a, bv, acc);
    }
#pragma unroll
    for (int g = 0; g < 8; ++g) {
      int ri = i0 + ((lane >> 4) << 3) + g;
      AWV[(size_t)(bz * nCS + ri) * nHD + h * nD + d0 + lm] = acc[g];
    }
  }
}

// ---------------------------------------------------------------------------
// Host-side launch. Workspace layout (floats):
//   K (bs*T*HD) | V (bs*T*HD) | Q (bs*cs*HD) | P (T*HD) | AWV (bs*cs*HD)
// ---------------------------------------------------------------------------
extern "C" void kernel_launch(void* const* d_in, const int* in_sizes, int n_in,
                              void* d_out, int out_size, void* d_ws, size_t ws_size,
                              hipStream_t stream) {
  (void)in_sizes; (void)n_in; (void)out_size; (void)ws_size;
  const float* input_ = (const float*)d_in[0];
  const float* pos    = (const float*)d_in[1];
  const float* memory = (const float*)d_in[2];
  const float* u      = (const float*)d_in[3];
  const float* v      = (const float*)d_in[4];
  const float* W_kv   = (const float*)d_in[5];
  const float* W_q    = (const float*)d_in[6];
  const float* W_p    = (const float*)d_in[7];
  const float* W_out  = (const float*)d_in[8];
  // d_in[9] (mask) ignored: recomputed as j <= i + PS inside the kernel.
  float* out = (float*)d_out;

  float* ws  = (float*)d_ws;
  float* Kws = ws;
  float* Vws = Kws + (size_t)nBS * nT * nHD;
  float* Qws = Vws + (size_t)nBS * nT * nHD;
  float* Pws = Qws + (size_t)nBS * nCS * nHD;
  float* AWV = Pws + (size_t)nT * nHD;

  // k_tfmd = [memory; input] @ W_kv[:, :1024]
  gemm_wmma_f32<<<dim3(nHD / 64, (nBS * nT) / 32), 32, 0, stream>>>(
      memory, input_, nT, nPS, nDIN, W_kv, 2 * nHD, Kws, nHD);
  // v_tfmd = [memory; input] @ W_kv[:, 1024:]
  gemm_wmma_f32<<<dim3(nHD / 64, (nBS * nT) / 32), 32, 0, stream>>>(
      memory, input_, nT, nPS, nDIN, W_kv + nHD, 2 * nHD, Vws, nHD);
  // q_tfmd = input @ W_q
  gemm_wmma_f32<<<dim3(nHD / 64, (nBS * nCS) / 32), 32, 0, stream>>>(
      input_, input_, nBS * nCS, nBS * nCS, nDIN, W_q, nHD, Qws, nHD);
  // p_tfmd = pos_embs @ W_p
  gemm_wmma_f32<<<dim3(nHD / 64, nT / 32), 32, 0, stream>>>(
      pos, pos, nT, nT, nDIN, W_p, nHD, Pws, nHD);

  // fused attention
  txl_attn_kernel<<<dim3(nCS / 16, nH, nBS), 128, 16 * nT * sizeof(float), stream>>>(
      Qws, Kws, Vws, Pws, u, v, AWV);

  // out = awv @ W_out
  gemm_wmma_f32<<<dim3(nDIN / 64, (nBS * nCS) / 32), 32, 0, stream>>>(
      AWV, AWV, nBS * nCS, nBS * nCS, nHD, W_out, nDIN, out, nDIN);
}